// phys_coESN_75299366633680
// MI455X (gfx1250) — compile-verified
//
#include <hip/hip_runtime.h>
#include <stddef.h>

typedef float v2f __attribute__((ext_vector_type(2)));
typedef float v8f __attribute__((ext_vector_type(8)));

constexpr int   Bdim = 512, Ldim = 512, Idim = 32, Hdim = 512;
constexpr float DT   = 0.042f;

// Native CDNA5 transcendental tanh (single V_TANH_F32) instead of the branchy
// OCML expansion. Fallback inline asm keeps the TRANS hazard rule satisfied
// with an explicit v_nop.
__device__ __forceinline__ float fast_tanh(float v) {
#if __has_builtin(__builtin_amdgcn_tanhf)
  return __builtin_amdgcn_tanhf(v);
#else
  float r;
  asm volatile("v_tanh_f32 %0, %1\n\tv_nop" : "=v"(r) : "v"(v));
  return r;
#endif
}

// Zero hy (ping buffer) and hz state each call — d_ws is poisoned by harness.
__global__ __launch_bounds__(256) void coesn_init(float* __restrict__ hy,
                                                  float* __restrict__ hz) {
  int i = blockIdx.x * blockDim.x + threadIdx.x;
  if (i < Bdim * Hdim) { hy[i] = 0.0f; hz[i] = 0.0f; }
}

// One full timestep. Workgroup j owns hidden columns [16j, 16j+16); wave w owns
// batch rows [16w, 16w+16). Both the h2h column block (GEMM1 B operand, shared
// by all 32 waves) and the intermediate T column block live in LDS. hy is
// double-buffered across launches; hz is column-owned and updated in place.
__global__ __launch_bounds__(1024) void coesn_step(
    const float* __restrict__ x,     // (B, L, I)
    const float* __restrict__ x2h,   // (I, H)
    const float* __restrict__ h2h,   // (H, H)
    const float* __restrict__ h2hT,  // (H, H)
    const float* __restrict__ bias,  // (H)
    const float* __restrict__ gma,   // (H)
    const float* __restrict__ eps,   // (H)
    const float* __restrict__ hy_in, // (B, H)  state at step t
    float* __restrict__ hy_out,      // (B, H)  state at step t+1
    float* __restrict__ hz,          // (B, H)  in-place
    float* __restrict__ out,         // (B, L, H) all_states
    float* __restrict__ hy_final,    // (B, H)  hy_f output
    int t)
{
  __shared__ float Wlds[Hdim][16];   // h2h[:, J]                : 32 KB
  __shared__ float Tlds[Hdim][16];   // tanh(hy@h2h+bias)[:, J]  : 32 KB

  const int lane  = threadIdx.x & 31;
  const int wave  = threadIdx.x >> 5;       // 0..31  -> M (row) tile
  const int col0  = blockIdx.x * 16;        // column block J
  const int row0  = wave * 16;
  const int n     = lane & 15;              // N for B/C/D, M for A (ISA layout)
  const int khalf = (lane >> 4) << 1;       // K sub-offset: 0 or 2
  const int mhi   = (lane >> 4) << 3;       // C/D row offset: 0 or 8

  // ---- stage shared GEMM1 B operand h2h[:, J] into LDS (once per step) ----
  #pragma unroll
  for (int i = threadIdx.x; i < Hdim * 16; i += 1024) {
    const int k = i >> 4, c = i & 15;
    Wlds[k][c] = h2h[(size_t)k * Hdim + col0 + c];
  }
  __syncthreads();

  // ---------- GEMM1: T = tanh(hy_in @ h2h + bias), tile (row0, col0) ----------
  v8f acc = {};
  {
    const float* arow = hy_in + (row0 + n) * Hdim;        // A: 16x4 f32 tiles
    #pragma unroll 4
    for (int k = 0; k < Hdim; k += 4) {
      v2f a = *(const v2f*)(arow + k + khalf);            // global_load_b64
      v2f b;
      b.x = Wlds[k + khalf][n];                           // conflict-free ds reads
      b.y = Wlds[k + khalf + 1][n];
      acc = __builtin_amdgcn_wmma_f32_16x16x4_f32(
          false, a, false, b, (short)0, acc, false, false);
    }
  }
  {
    const float bn = bias[col0 + n];
    #pragma unroll
    for (int v = 0; v < 8; ++v)
      Tlds[row0 + mhi + v][n] = fast_tanh(acc[v] + bn);   // v_tanh_f32
  }

  __syncthreads();   // T[:, J] complete in LDS

  // ---------- GEMM2: U = h2hT @ T   (K contracts over T's rows, from LDS) ----
  v8f acc2 = {};
  {
    const float* arow = h2hT + (row0 + n) * Hdim;
    #pragma unroll 4
    for (int k = 0; k < Hdim; k += 4) {
      v2f a = *(const v2f*)(arow + k + khalf);
      v2f b;
      b.x = Tlds[k + khalf][n];
      b.y = Tlds[k + khalf + 1][n];
      acc2 = __builtin_amdgcn_wmma_f32_16x16x4_f32(
          false, a, false, b, (short)0, acc2, false, false);
    }
  }

  // ---------- GEMM3: XW = x[:, t, :] @ x2h   (K = 32) -------------------------
  v8f acc3 = {};
  {
    const float* arow = x + (size_t)(row0 + n) * Ldim * Idim + (size_t)t * Idim;
    const float* bcol = x2h + col0 + n;
    #pragma unroll
    for (int k = 0; k < Idim; k += 4) {
      v2f a = *(const v2f*)(arow + k + khalf);
      v2f b;
      b.x = bcol[(size_t)(k + khalf) * Hdim];
      b.y = bcol[(size_t)(k + khalf + 1) * Hdim];
      acc3 = __builtin_amdgcn_wmma_f32_16x16x4_f32(
          false, a, false, b, (short)0, acc3, false, false);
    }
  }

  // ---------- oscillator update + streaming output -----------------------------
  const float g_n = gma[col0 + n];
  const float e_n = eps[col0 + n];
  #pragma unroll
  for (int v = 0; v < 8; ++v) {
    const int r = row0 + mhi + v;          // batch index
    const int c = col0 + n;                // hidden index
    const size_t sidx = (size_t)r * Hdim + c;
    const float hyv = hy_in[sidx];
    const float hzv = hz[sidx];
    const float hznew = hzv + DT * (acc3[v] - acc2[v] - g_n * hyv - e_n * hzv);
    const float hynew = hyv + DT * hznew;
    hz[sidx]     = hznew;
    hy_out[sidx] = hynew;
    // 512 MB stream: keep it out of L2 so h2h/h2h_T/hy stay resident
    __builtin_nontemporal_store(hynew,
        out + (size_t)r * Ldim * Hdim + (size_t)t * Hdim + c);
    if (t == Ldim - 1) hy_final[sidx] = hynew;
  }
}

extern "C" void kernel_launch(void* const* d_in, const int* in_sizes, int n_in,
                              void* d_out, int out_size, void* d_ws, size_t ws_size,
                              hipStream_t stream) {
  (void)in_sizes; (void)n_in; (void)out_size; (void)ws_size;
  const float* x    = (const float*)d_in[0];   // (B, L, I)
  const float* x2h  = (const float*)d_in[1];   // (I, H)
  const float* h2h  = (const float*)d_in[2];   // (H, H)
  const float* h2hT = (const float*)d_in[3];   // (H, H)
  const float* bias = (const float*)d_in[4];   // (H)
  const float* gma  = (const float*)d_in[5];   // (H)
  const float* eps  = (const float*)d_in[6];   // (H)

  float* out      = (float*)d_out;                              // (B, L, H)
  float* hy_final = out + (size_t)Bdim * Ldim * Hdim;           // (B, H)

  float* hyA = (float*)d_ws;                                    // ping
  float* hyB = hyA + (size_t)Bdim * Hdim;                       // pong
  float* hz  = hyB + (size_t)Bdim * Hdim;                       // hz state

  {
    const int nstate = Bdim * Hdim;
    coesn_init<<<(nstate + 255) / 256, 256, 0, stream>>>(hyA, hz);
  }

  const float* cur = hyA;
  float* nxt = hyB;
  for (int t = 0; t < Ldim; ++t) {
    coesn_step<<<dim3(Hdim / 16), dim3(1024), 0, stream>>>(
        x, x2h, h2h, h2hT, bias, gma, eps, cur, nxt, hz, out, hy_final, t);
    const float* tmp = nxt; nxt = (float*)cur; cur = tmp;
  }
}